// DigitCapsules_67912022884648
// MI455X (gfx1250) — compile-verified
//
#include <hip/hip_runtime.h>

// DigitCapsules dynamic routing, fully fused, u_hat recomputed via
// V_WMMA_F32_16X16X4_F32 (per-r GEMM: [160x8] W_r  @  [8x16] x-batch-tile).
//
// Pass p (p=0,1,2), one block = (16-batch tile, 72-row chunk):
//   per r: u_hat tile via WMMA -> (p>0: agreement with v_{p}, b update)
//          -> softmax over c -> s accumulation in registers.
//   deterministic LDS reduce -> s_part[chunk][b][c*16+o]
// squash kernel: sum chunks in fixed order, squash, write v (or d_out).

#define B_   256
#define R_   1152
#define C_   10
#define O_   16
#define I_   8
#define CO_  160            // C_*O_
#define NCHUNK 16
#define RPB  (R_/NCHUNK)    // 72 rows per block
#define WAVES 8
#define RPW  (RPB/WAVES)    // 9 rows per wave

typedef float v2f __attribute__((ext_vector_type(2)));
typedef float v8f __attribute__((ext_vector_type(8)));

__launch_bounds__(256)
__global__ void caps_route_pass(const float* __restrict__ x,      // [B,R,I]
                                const float* __restrict__ W,      // [R,C,O,I]
                                const float* __restrict__ b_src,  // [B,R,C]
                                float*       __restrict__ b_dst,  // [B,R,C] (pass==1)
                                const float* __restrict__ v_prev, // [B,C,O] (pass>0)
                                float*       __restrict__ s_part, // [NCHUNK,B,CO]
                                int pass)
{
    __shared__ float lds_s[4 * 16 * CO_];   // 4 wave-pair partials, 40 KB
    __shared__ float lds_v[16 * CO_];       // v tile for 16 batches, 10 KB

    const int tid   = threadIdx.x;
    const int wave  = tid >> 5;
    const int lane  = tid & 31;
    const int m     = lane & 15;    // batch within tile (N) == A-row index (M)
    const int kh    = lane >> 4;    // half selector: K pair + o-half + D M-half
    const int btile = blockIdx.x;   // 0..15
    const int chunk = blockIdx.y;   // 0..15
    const int b0    = btile * 16;
    const int b     = b0 + m;
    const int ohalf = 8 * kh;

    if (pass > 0) {
        for (int i = tid; i < 16 * CO_; i += 256)
            lds_v[i] = v_prev[(b0 + i / CO_) * CO_ + (i % CO_)];
    }
    __syncthreads();

    float sacc[C_][8];
    #pragma unroll
    for (int c = 0; c < C_; ++c)
        #pragma unroll
        for (int j = 0; j < 8; ++j) sacc[c][j] = 0.0f;

    for (int t = 0; t < RPW; ++t) {
        const int r = chunk * RPB + wave * RPW + t;

        // ---- B operand: x[b, r, K], K split across lane halves ----
        const float* xp = x + ((size_t)b * R_ + r) * I_;
        v2f xb0 = *(const v2f*)(xp + 2 * kh);       // K = {2kh, 2kh+1}
        v2f xb1 = *(const v2f*)(xp + 4 + 2 * kh);   // K = {4+2kh, 5+2kh}

        // ---- u_hat tile via WMMA: u[c][j] = u_hat[b, r, c, o=ohalf+j] ----
        float u[C_][8];
        const float* Wr = W + (size_t)r * CO_ * I_;
        #pragma unroll
        for (int c = 0; c < C_; ++c) {
            const float* Wp = Wr + (c * 16 + m) * I_;   // A row M = c*16+m
            v2f a0 = *(const v2f*)(Wp + 2 * kh);
            v2f a1 = *(const v2f*)(Wp + 4 + 2 * kh);
            v8f acc = {0.f,0.f,0.f,0.f,0.f,0.f,0.f,0.f};
            acc = __builtin_amdgcn_wmma_f32_16x16x4_f32(
                      false, a0, false, xb0, (short)0, acc, false, false);
            acc = __builtin_amdgcn_wmma_f32_16x16x4_f32(
                      false, a1, false, xb1, (short)0, acc, false, false);
            #pragma unroll
            for (int j = 0; j < 8; ++j) u[c][j] = acc[j];
        }

        // ---- routing logits ----
        float bij[C_];
        const float* bs = b_src + ((size_t)b * R_ + r) * C_;
        #pragma unroll
        for (int c = 0; c < C_; ++c) bij[c] = bs[c];

        if (pass > 0) {
            #pragma unroll
            for (int c = 0; c < C_; ++c) {
                const float* vv = &lds_v[m * CO_ + c * 16 + ohalf];
                float p = 0.0f;
                #pragma unroll
                for (int j = 0; j < 8; ++j) p += u[c][j] * vv[j];
                p += __shfl_xor(p, 16, 32);   // combine o-halves
                bij[c] += p;
            }
            if (pass == 1 && kh == 0) {
                float* bd = b_dst + ((size_t)b * R_ + r) * C_;
                #pragma unroll
                for (int c = 0; c < C_; ++c) bd[c] = bij[c];
            }
        }

        // ---- softmax over c (10 values, in registers) ----
        float mx = bij[0];
        #pragma unroll
        for (int c = 1; c < C_; ++c) mx = fmaxf(mx, bij[c]);
        float e[C_], sum = 0.0f;
        #pragma unroll
        for (int c = 0; c < C_; ++c) { e[c] = __expf(bij[c] - mx); sum += e[c]; }
        const float inv = 1.0f / sum;

        // ---- s accumulation ----
        #pragma unroll
        for (int c = 0; c < C_; ++c) {
            const float cij = e[c] * inv;
            #pragma unroll
            for (int j = 0; j < 8; ++j) sacc[c][j] += cij * u[c][j];
        }
    }

    // ---- deterministic block reduction: 8 waves -> 4 -> 1 ----
    const int pos = m * CO_ + /*c*16 + ohalf + j added below*/ 0;
    if (wave >= 4) {
        #pragma unroll
        for (int c = 0; c < C_; ++c)
            #pragma unroll
            for (int j = 0; j < 8; ++j)
                lds_s[(wave - 4) * (16 * CO_) + pos + c * 16 + ohalf + j] = sacc[c][j];
    }
    __syncthreads();
    if (wave < 4) {
        #pragma unroll
        for (int c = 0; c < C_; ++c)
            #pragma unroll
            for (int j = 0; j < 8; ++j) {
                const int q = wave * (16 * CO_) + pos + c * 16 + ohalf + j;
                lds_s[q] = sacc[c][j] + lds_s[q];   // wave w + wave w+4, single owner
            }
    }
    __syncthreads();
    for (int idx = tid; idx < 16 * CO_; idx += 256) {
        float s01 = lds_s[idx] + lds_s[16 * CO_ + idx];
        float s23 = lds_s[2 * 16 * CO_ + idx] + lds_s[3 * 16 * CO_ + idx];
        s_part[((size_t)chunk * B_ + b0 + idx / CO_) * CO_ + (idx % CO_)] = s01 + s23;
    }
}

__launch_bounds__(256)
__global__ void caps_squash(const float* __restrict__ s_part,  // [NCHUNK,B,CO]
                            float*       __restrict__ v_out)   // [B,C,O]
{
    const int idx = blockIdx.x * blockDim.x + threadIdx.x;  // (b*C + c)
    if (idx >= B_ * C_) return;
    const int b = idx / C_;
    const int c = idx % C_;

    float s[O_];
    #pragma unroll
    for (int o = 0; o < O_; ++o) s[o] = 0.0f;
    for (int ch = 0; ch < NCHUNK; ++ch) {           // fixed order: deterministic
        const float* sp = s_part + ((size_t)ch * B_ + b) * CO_ + c * O_;
        #pragma unroll
        for (int o = 0; o < O_; ++o) s[o] += sp[o];
    }

    float norm = 0.0f;
    #pragma unroll
    for (int o = 0; o < O_; ++o) norm += s[o] * s[o];
    const float scale = (norm / (1.0f + norm)) / sqrtf(norm + 1e-8f);

    float* vp = v_out + (size_t)b * CO_ + c * O_;
    #pragma unroll
    for (int o = 0; o < O_; ++o) vp[o] = scale * s[o];
}

extern "C" void kernel_launch(void* const* d_in, const int* in_sizes, int n_in,
                              void* d_out, int out_size, void* d_ws, size_t ws_size,
                              hipStream_t stream) {
    const float* x     = (const float*)d_in[0];   // [256,1152,8]
    const float* W     = (const float*)d_in[1];   // [1152,10,16,8]
    const float* binit = (const float*)d_in[2];   // [256,1152,10]
    float* out = (float*)d_out;                   // [256,10,16]

    float* ws   = (float*)d_ws;
    float* ws_b = ws;                                   // B*R*C   = 2,949,120 f
    float* ws_v = ws_b + (size_t)B_ * R_ * C_;          // B*C*O   = 40,960 f
    float* ws_s = ws_v + (size_t)B_ * C_ * O_;          // NCHUNK*B*CO = 655,360 f

    dim3 grid(16, NCHUNK), block(256);
    dim3 gsq((B_ * C_ + 255) / 256), bsq(256);

    // iter 1: softmax(b_init) -> s1 -> v1
    caps_route_pass<<<grid, block, 0, stream>>>(x, W, binit, nullptr, nullptr, ws_s, 0);
    caps_squash<<<gsq, bsq, 0, stream>>>(ws_s, ws_v);
    // iter 2: b1 = b_init + agree(v1); softmax -> s2 -> v2
    caps_route_pass<<<grid, block, 0, stream>>>(x, W, binit, ws_b, ws_v, ws_s, 1);
    caps_squash<<<gsq, bsq, 0, stream>>>(ws_s, ws_v);
    // iter 3: b2 = b1 + agree(v2); softmax -> s3 -> v3 (output)
    caps_route_pass<<<grid, block, 0, stream>>>(x, W, ws_b, nullptr, ws_v, ws_s, 2);
    caps_squash<<<gsq, bsq, 0, stream>>>(ws_s, out);
}